// Model_59313498358137
// MI455X (gfx1250) — compile-verified
//
#include <hip/hip_runtime.h>

#define NUM_HEADS 8
#define HEAD_DIM 32

typedef __attribute__((ext_vector_type(2))) float v2f;
typedef __attribute__((ext_vector_type(8))) float v8f;

// One block per destination node; wave w (of 8) handles head w.
// Stage up to 32 edges (cols + this head's weights) per lane with a single
// unconditional clamped load pair, then per 4-edge chunk build
//   A (16x4 f32): row 0 = weights, rows 1..15 = 0
//   B (4x16 f32): gathered neighbor features (coalesced 64B per half-wave)
// and chain D = A*B + C on the exact-f32 WMMA pipe (K=4).
__global__ __launch_bounds__(256) void gat_csr_gather_wmma(
    const int*   __restrict__ row_ptr,
    const int*   __restrict__ col_idx,
    const float* __restrict__ edge_weight,
    const float* __restrict__ node_feat,
    float*       __restrict__ out,
    int num_nodes)
{
    const int node = blockIdx.x;
    if (node >= num_nodes) return;

    const int h    = threadIdx.x >> 5;   // head = wave id (wave32)
    const int lane = threadIdx.x & 31;

    const int rbeg = row_ptr[node];
    const int rend = row_ptr[node + 1];

    v8f acc_lo = {};   // dims  0..15 : result row M=0 in element 0, lanes 0..15
    v8f acc_hi = {};   // dims 16..31

    const int  d      = lane & 15;
    const bool hihalf = lane >= 16;
    const int  voff   = h * HEAD_DIM + d;     // offset within a node's feature row
    const int  pbase  = hihalf ? 2 : 0;       // half-wave K-pair select for shuffles
    const bool row0   = (d == 0);             // lanes 0 and 16 carry A row 0

    for (int eb = rbeg; eb < rend; eb += 32) {
        // Stage 32 edges: clamped unconditional loads, weight masked past row end.
        const int el  = eb + lane;
        const int elc = min(el, rend - 1);
        const int   c_l = col_idx[elc];
        float w_l = edge_weight[elc * NUM_HEADS + h];
        w_l = (el < rend) ? w_l : 0.0f;

        const int nchunks = min((rend - eb + 3) >> 2, 8);
        for (int c = 0; c < nchunks; ++c) {
            const int p0 = 4 * c + pbase;

            // B column sources: lanes 0..15 get edges 4c,4c+1; lanes 16..31 get 4c+2,4c+3.
            const int cB0 = __shfl(c_l, p0);
            const int cB1 = __shfl(c_l, p0 + 1);

            // A row 0 weights: lane 0 -> K0/K1, lane 16 -> K2/K3, rest 0.
            const float wa0 = __shfl(w_l, p0);
            const float wa1 = __shfl(w_l, p0 + 1);
            v2f a;
            a.x = row0 ? wa0 : 0.0f;
            a.y = row0 ? wa1 : 0.0f;

            // 32-bit indices -> saddr + scaled voffset loads, offset:64 for +16 dims.
            const int i0 = cB0 * (NUM_HEADS * HEAD_DIM) + voff;
            const int i1 = cB1 * (NUM_HEADS * HEAD_DIM) + voff;
            v2f b_lo, b_hi;
            b_lo.x = node_feat[i0];       b_lo.y = node_feat[i1];       // dims 0..15
            b_hi.x = node_feat[i0 + 16];  b_hi.y = node_feat[i1 + 16];  // dims 16..31

            // 8 args: (neg_a, A, neg_b, B, c_mod, C, reuse_a, reuse_b)
            acc_lo = __builtin_amdgcn_wmma_f32_16x16x4_f32(
                false, a, false, b_lo, (short)0, acc_lo, false, false);
            acc_hi = __builtin_amdgcn_wmma_f32_16x16x4_f32(
                false, a, false, b_hi, (short)0, acc_hi, false, false);
        }
    }

    // D row M=0 -> vector element 0 on lanes 0..15.
    if (lane < 16) {
        float* o = out + ((size_t)node * NUM_HEADS + h) * HEAD_DIM;
        o[lane]      = acc_lo[0];
        o[lane + 16] = acc_hi[0];
    }
}

extern "C" void kernel_launch(void* const* d_in, const int* in_sizes, int n_in,
                              void* d_out, int out_size, void* d_ws, size_t ws_size,
                              hipStream_t stream) {
    const int*   row_ptr     = (const int*)d_in[0];
    const int*   col_idx     = (const int*)d_in[1];
    const float* edge_weight = (const float*)d_in[2];
    const float* node_feat   = (const float*)d_in[3];
    float*       out         = (float*)d_out;

    const int num_nodes = in_sizes[0] - 1;   // row_ptr has N+1 entries

    dim3 grid(num_nodes);
    dim3 block(256);   // 8 wave32 waves: one per head
    gat_csr_gather_wmma<<<grid, block, 0, stream>>>(
        row_ptr, col_idx, edge_weight, node_feat, out, num_nodes);
}